// RegionProposalNetwork_17239998726242
// MI455X (gfx1250) — compile-verified
//
#include <hip/hip_runtime.h>
#include <stdint.h>

#define NIMG 8
#define NA 3
#define NLVL 5
#define PRE 1000
#define POST 1000
#define KCAND (NLVL * PRE)   // 5000
#define KPAD 5120
#define SORTN 8192
#define IMGF 1344.0f
#define NMS_THR 0.7f
#define MINSZ 1e-3f
#define BCLAMP 4.135166556742356f   // log(1000/16)

__constant__ int d_H[NLVL]      = {336, 168, 84, 42, 21};
__constant__ int d_W[NLVL]      = {336, 168, 84, 42, 21};
__constant__ int d_OFF[NLVL + 1] = {0, 338688, 423360, 444528, 449820, 451143};

struct Ptrs {
    const float* obj[NLVL];
    const float* dl[NLVL];
    const float* anchors;
};

__device__ __forceinline__ uint32_t enc_f(float f) {
    uint32_t u = __float_as_uint(f);
    return (u & 0x80000000u) ? ~u : (u | 0x80000000u);
}
__device__ __forceinline__ float dec_k(uint32_t k) {
    uint32_t u = (k & 0x80000000u) ? (k ^ 0x80000000u) : ~k;
    return __uint_as_float(u);
}

// Decode anchor t for image n: BoxCoder.decode + clip + small-box validity.
__device__ __forceinline__ void decode_box(const Ptrs& P, int n, int t,
                                           float out[4], int* lvl_out, bool* valid_out) {
    int l = 0;
    while (l + 1 < NLVL && t >= d_OFF[l + 1]) ++l;
    int h = d_H[l], w = d_W[l];
    int hw = h * w;
    int tl = t - d_OFF[l];
    int a = tl % NA;
    int pos = tl / NA;
    int y = pos / w, x = pos % w;
    const float* dp = P.dl[l] + ((size_t)n * 4 * NA + (size_t)a * 4) * hw + (size_t)y * w + x;
    float dx = dp[0];
    float dy = dp[hw];
    float dw = dp[2 * hw];
    float dh = dp[3 * hw];
    const float* ap = P.anchors + 4 * (size_t)t;
    float ax1 = ap[0], ay1 = ap[1], ax2 = ap[2], ay2 = ap[3];
    float aw = ax2 - ax1, ah = ay2 - ay1;
    float acx = ax1 + 0.5f * aw, acy = ay1 + 0.5f * ah;
    dw = fminf(dw, BCLAMP);
    dh = fminf(dh, BCLAMP);
    float pcx = dx * aw + acx, pcy = dy * ah + acy;
    float pw = __expf(dw) * aw, ph = __expf(dh) * ah;
    float x1 = pcx - 0.5f * pw, y1 = pcy - 0.5f * ph;
    float x2 = pcx + 0.5f * pw, y2 = pcy + 0.5f * ph;
    x1 = fminf(fmaxf(x1, 0.0f), IMGF);
    y1 = fminf(fmaxf(y1, 0.0f), IMGF);
    x2 = fminf(fmaxf(x2, 0.0f), IMGF);
    y2 = fminf(fmaxf(y2, 0.0f), IMGF);
    out[0] = x1; out[1] = y1; out[2] = x2; out[3] = y2;
    *lvl_out = l;
    *valid_out = ((x2 - x1) >= MINSZ) && ((y2 - y1) >= MINSZ);
}

// ---------------------------------------------------------------------------
// Kernel A: per (image, level) exact top-1000 via 4-pass radix select + compact
// ---------------------------------------------------------------------------
__global__ __launch_bounds__(1024) void k_select(Ptrs P, int* cand_t, float* cand_s) {
    int bid = blockIdx.x;
    int n = bid / NLVL, l = bid % NLVL;
    int h = d_H[l], w = d_W[l];
    int hw = h * w;
    int len = NA * hw;
    const float* src = P.obj[l] + (size_t)n * len;
    int tid = threadIdx.x;

    __shared__ int hist[256];
    __shared__ uint32_t sh_prefix;
    __shared__ int sh_want;
    __shared__ int sh_cnt;

    if (tid == 0) { sh_prefix = 0u; sh_want = PRE; }
    __syncthreads();

    for (int pass = 0; pass < 4; ++pass) {
        for (int b = tid; b < 256; b += blockDim.x) hist[b] = 0;
        __syncthreads();
        uint32_t pfx = sh_prefix;
        int shiftHi = 32 - 8 * pass;
        int shiftB = 24 - 8 * pass;
        for (int e = tid; e < len; e += blockDim.x) {
            uint32_t key = enc_f(src[e]);
            if (pass == 0 || ((key >> shiftHi) == pfx))
                atomicAdd(&hist[(key >> shiftB) & 255], 1);
        }
        __syncthreads();
        if (tid == 0) {
            int want = sh_want;
            int sel = 0;
            for (int b = 255; b >= 0; --b) {
                if (want <= hist[b]) { sel = b; break; }
                want -= hist[b];
            }
            sh_want = want;
            sh_prefix = (sh_prefix << 8) | (uint32_t)sel;
        }
        __syncthreads();
    }
    uint32_t thresh = sh_prefix;   // key of exact 1000th-largest score
    if (tid == 0) sh_cnt = 0;
    __syncthreads();
    for (int e = tid; e < len; e += blockDim.x) {
        float f = src[e];
        if (enc_f(f) >= thresh) {
            int p = atomicAdd(&sh_cnt, 1);
            if (p < PRE) {
                int a = e / hw;
                int pos = e % hw;
                int t = d_OFF[l] + pos * NA + a;   // permute_and_flatten index
                int slot = n * KPAD + l * PRE + p;
                cand_t[slot] = t;
                cand_s[slot] = f;
            }
        }
    }
}

// ---------------------------------------------------------------------------
// Kernel B: build 64-bit sort keys (score desc, anchor-index tiebreak, invalid->0)
// ---------------------------------------------------------------------------
__global__ void k_keys(Ptrs P, const int* cand_t, const float* cand_s, uint64_t* key64) {
    int i = blockIdx.x * blockDim.x + threadIdx.x;
    if (i >= NIMG * KCAND) return;
    int n = i / KCAND, slot = i % KCAND;
    int t = cand_t[n * KPAD + slot];
    float s = cand_s[n * KPAD + slot];
    float box[4]; int l; bool valid;
    decode_box(P, n, t, box, &l, &valid);
    uint32_t sk = valid ? enc_f(s) : 0u;
    key64[(size_t)n * SORTN + slot] = ((uint64_t)sk << 32) | (uint32_t)(~(uint32_t)t);
}

// ---------------------------------------------------------------------------
// Kernel C: per-image bitonic sort of 8192 keys in 64 KB LDS (descending)
// ---------------------------------------------------------------------------
__global__ __launch_bounds__(1024) void k_sort(uint64_t* key64) {
    __shared__ uint64_t sk[SORTN];
    int n = blockIdx.x, tid = threadIdx.x;
    uint64_t* base = key64 + (size_t)n * SORTN;
    for (int i = tid; i < SORTN; i += 1024) sk[i] = (i < KCAND) ? base[i] : 0ull;
    __syncthreads();
    for (int size = 2; size <= SORTN; size <<= 1) {
        for (int stride = size >> 1; stride > 0; stride >>= 1) {
            for (int t = tid; t < SORTN / 2; t += 1024) {
                int lo = 2 * t - (t & (stride - 1));
                int hi = lo + stride;
                bool desc = ((lo & size) == 0);
                uint64_t a = sk[lo], b = sk[hi];
                if (desc ? (a < b) : (a > b)) { sk[lo] = b; sk[hi] = a; }
            }
            __syncthreads();
        }
    }
    for (int i = tid; i < KCAND; i += 1024) base[i] = sk[i];
}

// ---------------------------------------------------------------------------
// Kernel D: re-decode boxes at sorted (deterministic) positions
// ---------------------------------------------------------------------------
__global__ void k_decode2(Ptrs P, const uint64_t* key64,
                          float4* nbox, int* nmeta, float* nscore) {
    int i = blockIdx.x * blockDim.x + threadIdx.x;
    if (i >= NIMG * KCAND) return;
    int n = i / KCAND, r = i % KCAND;
    uint64_t kk = key64[(size_t)n * SORTN + r];
    uint32_t sk = (uint32_t)(kk >> 32);
    int t = (int)(~(uint32_t)kk);
    float box[4]; int l; bool valid;
    decode_box(P, n, t, box, &l, &valid);
    bool ok = valid && (sk != 0u);
    size_t o = (size_t)n * KPAD + r;
    nbox[o] = make_float4(box[0], box[1], box[2], box[3]);
    nmeta[o] = ok ? (l + 1) : 0;     // 0 == invalid
    nscore[o] = dec_k(sk);
}

// ---------------------------------------------------------------------------
// Kernel E: per-image NMS entirely in LDS (80 KB box tile staged with the
// CDNA5 async-to-LDS path), then emit (POST,5) output rows in score order.
// ---------------------------------------------------------------------------
__global__ __launch_bounds__(1024) void k_nms(const float4* nbox, const int* nmeta,
                                              const float* nscore, float* out) {
    __shared__ float4 sbox[KCAND];           // 80 KB — needs CDNA5's 320 KB LDS
    __shared__ unsigned char smeta[KCAND];
    __shared__ unsigned char ssup[KCAND];
    __shared__ unsigned char skeep[KCAND];

    int n = blockIdx.x, tid = threadIdx.x;

    // Zero the output region for this image (harness poisons d_out).
    for (int i = tid; i < POST * 5; i += 1024) out[(size_t)n * POST * 5 + i] = 0.0f;

    // Async-stage the sorted boxes into LDS (bypasses VGPRs).
    const float4* gb = nbox + (size_t)n * KPAD;
    for (int j = tid; j < KCAND; j += 1024) {
        uint32_t laddr = (uint32_t)(uintptr_t)(&sbox[j]);
        const float4* ga = gb + j;
        asm volatile("global_load_async_to_lds_b128 %0, %1, off"
                     :: "v"(laddr), "v"(ga) : "memory");
    }
    for (int j = tid; j < KCAND; j += 1024) {
        smeta[j] = (unsigned char)nmeta[(size_t)n * KPAD + j];
        ssup[j] = 0;
        skeep[j] = 0;
    }
    asm volatile("s_wait_asynccnt 0" ::: "memory");
    __syncthreads();

    // Sequential NMS: serial over i, parallel suppression over j>i.
    for (int i = 0; i < KCAND; ++i) {
        if (tid == 0) skeep[i] = (smeta[i] != 0 && ssup[i] == 0) ? 1 : 0;
        __syncthreads();
        if (skeep[i]) {
            float4 bi = sbox[i];
            unsigned char li = smeta[i];
            float areai = (bi.z - bi.x) * (bi.w - bi.y);
            for (int j = i + 1 + tid; j < KCAND; j += 1024) {
                if (ssup[j] || smeta[j] != li) continue;  // cross-level IoU == 0
                float4 bj = sbox[j];
                float xx1 = fmaxf(bi.x, bj.x), yy1 = fmaxf(bi.y, bj.y);
                float xx2 = fminf(bi.z, bj.z), yy2 = fminf(bi.w, bj.w);
                float iw = fmaxf(xx2 - xx1, 0.0f), ih = fmaxf(yy2 - yy1, 0.0f);
                float inter = iw * ih;
                float areaj = (bj.z - bj.x) * (bj.w - bj.y);
                float iou = inter / (areai + areaj - inter);
                if (iou > NMS_THR) ssup[j] = 1;
            }
        }
        __syncthreads();
    }

    // Emit kept rows (already score-descending) — serial, deterministic.
    if (tid == 0) {
        int cnt = 0;
        for (int i = 0; i < KCAND && cnt < POST; ++i) {
            if (skeep[i]) {
                float4 b = sbox[i];
                size_t o = (size_t)n * POST * 5 + (size_t)cnt * 5;
                out[o + 0] = b.x; out[o + 1] = b.y;
                out[o + 2] = b.z; out[o + 3] = b.w;
                out[o + 4] = nscore[(size_t)n * KPAD + i];
                ++cnt;
            }
        }
    }
}

extern "C" void kernel_launch(void* const* d_in, const int* in_sizes, int n_in,
                              void* d_out, int out_size, void* d_ws, size_t ws_size,
                              hipStream_t stream) {
    Ptrs P;
    for (int l = 0; l < NLVL; ++l) {
        P.obj[l] = (const float*)d_in[2 * l];       // obj0,delta0,obj1,delta1,...
        P.dl[l]  = (const float*)d_in[2 * l + 1];
    }
    P.anchors = (const float*)d_in[10];

    char* ws = (char*)d_ws;
    size_t off = 0;
    auto alloc = [&](size_t sz) -> void* {
        off = (off + 255) & ~(size_t)255;
        void* p = ws + off;
        off += sz;
        return p;
    };
    int*      cand_t = (int*)alloc((size_t)NIMG * KPAD * 4);
    float*    cand_s = (float*)alloc((size_t)NIMG * KPAD * 4);
    uint64_t* key64  = (uint64_t*)alloc((size_t)NIMG * SORTN * 8);
    float4*   nbox   = (float4*)alloc((size_t)NIMG * KPAD * 16);
    int*      nmeta  = (int*)alloc((size_t)NIMG * KPAD * 4);
    float*    nscore = (float*)alloc((size_t)NIMG * KPAD * 4);

    k_select<<<NIMG * NLVL, 1024, 0, stream>>>(P, cand_t, cand_s);
    int tot = NIMG * KCAND;
    k_keys<<<(tot + 255) / 256, 256, 0, stream>>>(P, cand_t, cand_s, key64);
    k_sort<<<NIMG, 1024, 0, stream>>>(key64);
    k_decode2<<<(tot + 255) / 256, 256, 0, stream>>>(P, key64, nbox, nmeta, nscore);
    k_nms<<<NIMG, 1024, 0, stream>>>(nbox, nmeta, nscore, (float*)d_out);
}